// APSDG_30124900614677
// MI455X (gfx1250) — compile-verified
//
#include <hip/hip_runtime.h>
#include <hip/hip_bf16.h>
#include <math.h>

typedef __attribute__((ext_vector_type(2))) float v2f;
typedef __attribute__((ext_vector_type(8))) float v8f;
typedef __attribute__((address_space(3))) float lds_f;

#define NN_NODES 50000
#define EMBD 384
#define SUBD 128
#define HISTW 5

static __device__ __forceinline__ unsigned lds_addr(const void* p) {
  return (unsigned)(unsigned long long)(lds_f*)p;
}

// =====================================================================
// FP32 WMMA GEMM:  C[M,Nc] = A[M,K](lda) @ B[K,Nc](ldb)  (+bias[n]) (*rowscale[m])
// TRANSB: B element (k,n) read from B[n*ldb + k]  (row-major B^T view)
// V_WMMA_F32_16X16X4_F32 compute + GLOBAL_LOAD_ASYNC_TO_LDS double-buffered
// staging (ASYNCcnt-tracked; no VGPR round-trip for tile data).
// Block: 256 thr = 8 waves; tile 128(M) x 128(N); wave tile 32x64.
// Requires: K % 32 == 0, Nc % 128 == 0 (holds for all call sites).
// =====================================================================
template<bool TRANSB>
__global__ __launch_bounds__(256) void k_gemm(
    const float* __restrict__ A, int lda,
    const float* __restrict__ B, int ldb,
    const float* __restrict__ bias,
    const float* __restrict__ rowscale,
    float* __restrict__ C, int ldc,
    int M, int Nc, int K)
{
  constexpr int BM = 128, BN = 128, KC = 32;
  constexpr int SAS = KC + 4;   // 36: rows 16B-aligned, conflict-free strided frag reads
  constexpr int SBS = BN + 4;   // 132: row base 528B (16B-aligned)
  __shared__ float sA[2][BM][SAS];
  __shared__ float sB[2][KC][SBS];

  const int tid  = threadIdx.x;
  const int lane = tid & 31, wave = tid >> 5;
  const int wm = wave >> 1, wn = wave & 1;      // 4x2 wave grid
  const int lm = lane & 15, lh = lane >> 4;     // lane row/col, K-half
  const int bm = blockIdx.y * BM, bn = blockIdx.x * BN;
  const int row_w = bm + wm * 32, col_w = bn + wn * 64;

  const unsigned lbaseA = lds_addr(&sA[0][0][0]);
  const unsigned lbaseB = lds_addr(&sB[0][0][0]);

  // async-stage one K-chunk into LDS buffer `buf`
  auto stage = [&](int buf, int k0) {
    // A tile 128x32 as b128 (4 dwords/lane/issue); clamp tail rows (masked at store)
    #pragma unroll
    for (int t = 0; t < (BM * KC / 4) / 256; ++t) {
      int i = tid + t * 256;
      int r = i >> 3, c4 = (i & 7) << 2;
      int gr = bm + r; if (gr >= M) gr = M - 1;
      unsigned goff = (unsigned)(((size_t)gr * lda + (size_t)(k0 + c4)) * 4u);
      unsigned loff = lbaseA + (unsigned)((buf * BM * SAS + r * SAS + c4) * 4);
      asm volatile("global_load_async_to_lds_b128 %0, %1, %2"
                   :: "v"(loff), "v"(goff), "s"(A) : "memory");
    }
    if (TRANSB) {
      // B^T: gather per-element (k-contiguous in memory, n-contiguous in LDS)
      #pragma unroll
      for (int t = 0; t < (KC * BN) / 256; ++t) {
        int i = tid + t * 256;
        int kk = i >> 7, c = i & 127;
        unsigned goff = (unsigned)(((size_t)(bn + c) * ldb + (size_t)(k0 + kk)) * 4u);
        unsigned loff = lbaseB + (unsigned)((buf * KC * SBS + kk * SBS + c) * 4);
        asm volatile("global_load_async_to_lds_b32 %0, %1, %2"
                     :: "v"(loff), "v"(goff), "s"(B) : "memory");
      }
    } else {
      // B rows contiguous: b128
      #pragma unroll
      for (int t = 0; t < (KC * BN / 4) / 256; ++t) {
        int i = tid + t * 256;
        int kk = i >> 5, c4 = (i & 31) << 2;
        unsigned goff = (unsigned)(((size_t)(k0 + kk) * ldb + (size_t)(bn + c4)) * 4u);
        unsigned loff = lbaseB + (unsigned)((buf * KC * SBS + kk * SBS + c4) * 4);
        asm volatile("global_load_async_to_lds_b128 %0, %1, %2"
                     :: "v"(loff), "v"(goff), "s"(B) : "memory");
      }
    }
  };

  v8f acc[2][4] = {};

  const int nch = K / KC;
  stage(0, 0);
  for (int ch = 0; ch < nch; ++ch) {
    // own async writes landed in LDS; barrier => everyone's landed AND all waves
    // are done reading the buffer the next prefetch will overwrite
    asm volatile("s_wait_asynccnt 0x0" ::: "memory");
    __syncthreads();
    if (ch + 1 < nch) stage((ch + 1) & 1, (ch + 1) * KC);   // overlaps with compute below

    const int buf = ch & 1;
    #pragma unroll
    for (int kk = 0; kk < KC; kk += 4) {
      v2f af[2], bf[4];
      #pragma unroll
      for (int i = 0; i < 2; i++) {
        int m = wm * 32 + i * 16 + lm;
        af[i][0] = sA[buf][m][kk + 2 * lh];       // VGPR0: K=0 / K=2
        af[i][1] = sA[buf][m][kk + 2 * lh + 1];   // VGPR1: K=1 / K=3
      }
      #pragma unroll
      for (int j = 0; j < 4; j++) {
        int n = wn * 64 + j * 16 + lm;
        bf[j][0] = sB[buf][kk + 2 * lh][n];
        bf[j][1] = sB[buf][kk + 2 * lh + 1][n];
      }
      #pragma unroll
      for (int i = 0; i < 2; i++)
        #pragma unroll
        for (int j = 0; j < 4; j++)
          acc[i][j] = __builtin_amdgcn_wmma_f32_16x16x4_f32(
              false, af[i], false, bf[j], (short)0, acc[i][j], false, false);
    }
  }

  // C/D layout: VGPR r -> M = r + 8*lh ; N = lane%16
  #pragma unroll
  for (int i = 0; i < 2; i++) {
    #pragma unroll
    for (int j = 0; j < 4; j++) {
      int n = col_w + j * 16 + lm;
      if (n < Nc) {
        float bval = bias ? bias[n] : 0.f;
        #pragma unroll
        for (int r = 0; r < 8; r++) {
          int m = row_w + i * 16 + r + 8 * lh;
          if (m < M) {
            float v = acc[i][j][r] + bval;
            if (rowscale) v *= rowscale[m];
            C[(size_t)m * ldc + n] = v;
          }
        }
      }
    }
  }
}

// =====================================================================
// Elementwise / graph kernels
// =====================================================================
__global__ void k_zero(float* p, long long n) {
  long long i = (long long)blockIdx.x * blockDim.x + threadIdx.x;
  long long st = (long long)gridDim.x * blockDim.x;
  for (; i < n; i += st) p[i] = 0.f;
}

__global__ void k_copy(float* dst, const float* src, long long n) {
  long long i = (long long)blockIdx.x * blockDim.x + threadIdx.x;
  long long st = (long long)gridDim.x * blockDim.x;
  for (; i < n; i += st) dst[i] = src[i];
}

__global__ void k_degree(const int* __restrict__ src, const int* __restrict__ dst,
                         float* dego, float* degi, int E) {
  int i = blockIdx.x * blockDim.x + threadIdx.x;
  if (i < E) {
    atomicAdd(&dego[src[i]], 1.f);
    atomicAdd(&degi[dst[i]], 1.f);
  }
}

__global__ void k_deg_fin(const float* dego, const float* degi,
                          float* iso, float* isi, float* iin, int n) {
  int i = blockIdx.x * blockDim.x + threadIdx.x;
  if (i < n) {
    float o = fmaxf(dego[i], 1.f), d = fmaxf(degi[i], 1.f);
    iso[i] = 1.f / sqrtf(o);
    isi[i] = 1.f / sqrtf(d);
    iin[i] = 1.f / d;
  }
}

// per-node: logmap0 of b-slice -> pre[:,0:128], l2norm of s-slice -> pre[:,128:256]
__global__ __launch_bounds__(128) void k_pre(const float* __restrict__ cur,
                                             const float* __restrict__ curv,
                                             float* __restrict__ pre) {
  int n = blockIdx.x, c = threadIdx.x;
  __shared__ float red[128];
  size_t base = (size_t)n * EMBD;

  float xb = cur[base + SUBD + c];
  red[c] = xb * xb; __syncthreads();
  for (int s = 64; s > 0; s >>= 1) { if (c < s) red[c] += red[c + s]; __syncthreads(); }
  float nrm = sqrtf(red[0]); __syncthreads();
  float sc = sqrtf(fabsf(curv[0]));
  float sn = sc * nrm;
  float snc = fminf(fmaxf(sn, 0.f), 1.f - 1e-5f);
  float scale = atanhf(snc) / fmaxf(sn, 1e-5f);
  pre[(size_t)n * 256 + c] = scale * xb;

  float xs = cur[base + 2 * SUBD + c];
  red[c] = xs * xs; __syncthreads();
  for (int s = 64; s > 0; s >>= 1) { if (c < s) red[c] += red[c + s]; __syncthreads(); }
  float nrm2 = sqrtf(red[0]); __syncthreads();
  pre[(size_t)n * 256 + 128 + c] = xs / fmaxf(nrm2, 1e-12f);
}

// agg[dst] += msg[src]; float4 loads, 4 atomics per thread
__global__ void k_scatter(const float* __restrict__ msg, const int* __restrict__ src,
                          const int* __restrict__ dst, float* __restrict__ agg,
                          long long total4) {
  long long i = (long long)blockIdx.x * blockDim.x + threadIdx.x;
  if (i < total4) {
    int e = (int)(i / (EMBD / 4)), c4 = (int)(i % (EMBD / 4)) * 4;
    const float4 v = *(const float4*)&msg[(size_t)src[e] * EMBD + c4];
    float* a = &agg[(size_t)dst[e] * EMBD + c4];
    atomicAdd(a + 0, v.x);
    atomicAdd(a + 1, v.y);
    atomicAdd(a + 2, v.z);
    atomicAdd(a + 3, v.w);
  }
}

// per-node layer epilogue: e = leaky(agg*isi + e_b); b = expmap0(agg*iin); s = l2norm(agg*iin)
__global__ __launch_bounds__(128) void k_layer_epi(
    const float* __restrict__ agg, const float* __restrict__ isi,
    const float* __restrict__ iin, const float* __restrict__ e_b,
    const float* __restrict__ curv, float* __restrict__ cur) {
  int n = blockIdx.x, c = threadIdx.x;
  __shared__ float red[128];
  size_t base = (size_t)n * EMBD;

  float v = agg[base + c] * isi[n] + e_b[c];
  cur[base + c] = v >= 0.f ? v : 0.2f * v;

  float ub = agg[base + SUBD + c] * iin[n];
  red[c] = ub * ub; __syncthreads();
  for (int s = 64; s > 0; s >>= 1) { if (c < s) red[c] += red[c + s]; __syncthreads(); }
  float nrm = sqrtf(red[0]); __syncthreads();
  float sc = sqrtf(fabsf(curv[0]));
  float sn = sc * nrm;
  cur[base + SUBD + c] = (tanhf(sn) / fmaxf(sn, 1e-5f)) * ub;

  float us = agg[base + 2 * SUBD + c] * iin[n];
  red[c] = us * us; __syncthreads();
  for (int s = 64; s > 0; s >>= 1) { if (c < s) red[c] += red[c + s]; __syncthreads(); }
  float nrm2 = sqrtf(red[0]); __syncthreads();
  cur[base + 2 * SUBD + c] = us / fmaxf(nrm2, 1e-12f);
}

// per-node attention: scores from qk, softmax over W=5, hctx = sum_w attn*history
__global__ __launch_bounds__(128) void k_attn(const float* __restrict__ qk,
                                              const float* __restrict__ hist,
                                              float* __restrict__ hctx) {
  int n = blockIdx.x, c = threadIdx.x;
  __shared__ float sh[HISTW * EMBD];
  __shared__ float red[128];
  __shared__ float sw[HISTW];

  for (int w = 0; w < HISTW; ++w)
    for (int d = c; d < EMBD; d += 128)
      sh[w * EMBD + d] = hist[((size_t)w * NN_NODES + n) * EMBD + d];
  float qv[3];
  #pragma unroll
  for (int j = 0; j < 3; j++) qv[j] = qk[(size_t)n * EMBD + j * 128 + c];
  __syncthreads();

  for (int w = 0; w < HISTW; ++w) {
    float p = 0.f;
    #pragma unroll
    for (int j = 0; j < 3; j++) p += sh[w * EMBD + j * 128 + c] * qv[j];
    red[c] = p; __syncthreads();
    for (int s = 64; s > 0; s >>= 1) { if (c < s) red[c] += red[c + s]; __syncthreads(); }
    if (c == 0) sw[w] = red[0] * 0.05103103630f;  // 384^-0.5
    __syncthreads();
  }
  float mx = sw[0];
  for (int w = 1; w < HISTW; w++) mx = fmaxf(mx, sw[w]);
  float a[HISTW], ssum = 0.f;
  for (int w = 0; w < HISTW; w++) { a[w] = expf(sw[w] - mx); ssum += a[w]; }
  float inv = 1.f / ssum;
  #pragma unroll
  for (int j = 0; j < 3; j++) {
    float accv = 0.f;
    for (int w = 0; w < HISTW; w++) accv += a[w] * sh[w * EMBD + j * 128 + c];
    hctx[(size_t)n * EMBD + j * 128 + c] = accv * inv;
  }
}

__global__ void k_gru(const float* __restrict__ gi, const float* __restrict__ gh,
                      const float* __restrict__ ctx, float* __restrict__ out,
                      long long total) {
  long long i = (long long)blockIdx.x * blockDim.x + threadIdx.x;
  if (i < total) {
    size_t nn = (size_t)(i / EMBD), d = (size_t)(i % EMBD);
    size_t b = nn * (3 * EMBD);
    float r = 1.f / (1.f + expf(-(gi[b + d] + gh[b + d])));
    float z = 1.f / (1.f + expf(-(gi[b + EMBD + d] + gh[b + EMBD + d])));
    float nm = tanhf(gi[b + 2 * EMBD + d] + r * gh[b + 2 * EMBD + d]);
    out[i] = (1.f - z) * nm + z * ctx[i];
  }
}

// =====================================================================
// Launcher
// =====================================================================
static inline void launch_gemm(bool transB,
                               const float* A, int lda, const float* B, int ldb,
                               const float* bias, const float* rowscale,
                               float* C, int ldc, int M, int Nc, int K,
                               hipStream_t s) {
  dim3 g((Nc + 127) / 128, (M + 127) / 128);
  if (transB)
    k_gemm<true><<<g, 256, 0, s>>>(A, lda, B, ldb, bias, rowscale, C, ldc, M, Nc, K);
  else
    k_gemm<false><<<g, 256, 0, s>>>(A, lda, B, ldb, bias, rowscale, C, ldc, M, Nc, K);
}

extern "C" void kernel_launch(void* const* d_in, const int* in_sizes, int n_in,
                              void* d_out, int out_size, void* d_ws, size_t ws_size,
                              hipStream_t stream) {
  const int*   src      = (const int*)d_in[0];
  const int*   dst      = (const int*)d_in[1];
  const float* history  = (const float*)d_in[2];
  const float* node_emb = (const float*)d_in[3];
  const float* e_W      = (const float*)d_in[4];
  const float* e_b      = (const float*)d_in[5];
  const float* b_W      = (const float*)d_in[6];
  const float* b_bias   = (const float*)d_in[7];
  const float* s_W      = (const float*)d_in[8];
  const float* s_bias   = (const float*)d_in[9];
  const float* q_W      = (const float*)d_in[10];
  const float* q_b      = (const float*)d_in[11];
  const float* in_w     = (const float*)d_in[12];
  const float* in_b     = (const float*)d_in[13];
  const float* out_w    = (const float*)d_in[14];
  const float* out_b    = (const float*)d_in[15];
  const float* gru_wih  = (const float*)d_in[16];
  const float* gru_bih  = (const float*)d_in[17];
  const float* gru_whh  = (const float*)d_in[18];
  const float* gru_bhh  = (const float*)d_in[19];
  const float* curv     = (const float*)d_in[20];
  const int E = in_sizes[0];
  float* out = (float*)d_out;

  const size_t N = NN_NODES;
  float* ws = (float*)d_ws;
  size_t o = 0;
  float* cur = ws + o; o += N * EMBD;
  float* msg = ws + o; o += N * EMBD;       // reused: q1, qk, ctx0
  float* agg = ws + o; o += N * EMBD;       // reused: q, hctx
  float* pre = ws + o; o += N * 256;
  float* ctx = ws + o; o += N * EMBD;
  float* gi  = ws + o; o += N * (3 * EMBD);
  float* gh  = ws + o; o += N * (3 * EMBD);
  float* dego = ws + o; o += N;
  float* degi = ws + o; o += N;
  float* iso  = ws + o; o += N;
  float* isi  = ws + o; o += N;
  float* iin  = ws + o; o += N;

  // degrees
  k_zero<<<2048, 256, 0, stream>>>(dego, (long long)2 * N);
  k_degree<<<(E + 255) / 256, 256, 0, stream>>>(src, dst, dego, degi, E);
  k_deg_fin<<<((int)N + 255) / 256, 256, 0, stream>>>(dego, degi, iso, isi, iin, (int)N);

  // cur = node_emb
  k_copy<<<4096, 256, 0, stream>>>(cur, node_emb, (long long)N * EMBD);

  const long long scat_total4 = (long long)E * (EMBD / 4);
  const int scat_blocks = (int)((scat_total4 + 255) / 256);

  // ---- graph layers ----
  for (int l = 0; l < 2; ++l) {
    k_pre<<<(int)N, 128, 0, stream>>>(cur, curv, pre);
    // e: msg[:,0:128] = (e @ e_W[l]) * inv_sqrt_out
    launch_gemm(false, cur, EMBD, e_W + (size_t)l * SUBD * SUBD, SUBD,
                nullptr, iso, msg, EMBD, (int)N, SUBD, SUBD, stream);
    // b: msg[:,128:256] = logmap0(b) @ b_W[l] + b_bias[l]
    launch_gemm(false, pre, 256, b_W + (size_t)l * SUBD * SUBD, SUBD,
                b_bias + (size_t)l * SUBD, nullptr, msg + SUBD, EMBD, (int)N, SUBD, SUBD, stream);
    // s: msg[:,256:384] = l2norm(s) @ s_W[l] + s_bias[l]
    launch_gemm(false, pre + 128, 256, s_W + (size_t)l * SUBD * SUBD, SUBD,
                s_bias + (size_t)l * SUBD, nullptr, msg + 2 * SUBD, EMBD, (int)N, SUBD, SUBD, stream);

    k_zero<<<4096, 256, 0, stream>>>(agg, (long long)N * EMBD);
    k_scatter<<<scat_blocks, 256, 0, stream>>>(msg, src, dst, agg, scat_total4);
    k_layer_epi<<<(int)N, 128, 0, stream>>>(agg, isi, iin, e_b + (size_t)l * SUBD, curv, cur);
  }

  // ---- attention (k/v buffers eliminated algebraically) ----
  // q1 (msg) = cur @ q_W + q_b
  launch_gemm(false, cur, EMBD, q_W, EMBD, q_b, nullptr, msg, EMBD, (int)N, EMBD, EMBD, stream);
  // q (agg) = q1 @ in_w[:, :D] + in_b[:D]
  launch_gemm(false, msg, EMBD, in_w, 3 * EMBD, in_b, nullptr, agg, EMBD, (int)N, EMBD, EMBD, stream);
  // qk (msg) = q @ Wk^T   (Wk = in_w[:, D:2D]); q·bk is constant over w -> cancels in softmax
  launch_gemm(true, agg, EMBD, in_w + EMBD, 3 * EMBD, nullptr, nullptr, msg, EMBD, (int)N, EMBD, EMBD, stream);
  // scores + softmax + history-space context (hctx in agg)
  k_attn<<<(int)N, 128, 0, stream>>>(msg, history, agg);
  // ctx0 (msg) = hctx @ Wv + bv   (sum attn = 1 so bv passes through)
  launch_gemm(false, agg, EMBD, in_w + 2 * EMBD, 3 * EMBD, in_b + 2 * EMBD, nullptr,
              msg, EMBD, (int)N, EMBD, EMBD, stream);
  // ctx = ctx0 @ out_w + out_b
  launch_gemm(false, msg, EMBD, out_w, EMBD, out_b, nullptr, ctx, EMBD, (int)N, EMBD, EMBD, stream);

  // ---- GRU ----
  launch_gemm(false, cur, EMBD, gru_wih, 3 * EMBD, gru_bih, nullptr, gi, 3 * EMBD,
              (int)N, 3 * EMBD, EMBD, stream);
  launch_gemm(false, ctx, EMBD, gru_whh, 3 * EMBD, gru_bhh, nullptr, gh, 3 * EMBD,
              (int)N, 3 * EMBD, EMBD, stream);
  k_gru<<<(int)((N * EMBD + 255) / 256), 256, 0, stream>>>(gi, gh, ctx, out, (long long)(N * EMBD));
}